// GraphMemory_19567871001099
// MI455X (gfx1250) — compile-verified
//
#include <hip/hip_runtime.h>

// ---------------------------------------------------------------------------
// Problem constants (from setup_inputs): B=4, P=64, D=768, N=2048, R=12, k=16
// ---------------------------------------------------------------------------
#define BB 4
#define PP 64
#define MM (BB * PP)   // 256 query rows
#define DD 768
#define NN 2048
#define RR 12
#define KK 16

typedef __attribute__((ext_vector_type(2))) float v2f;
typedef __attribute__((ext_vector_type(4))) float v4f;
typedef __attribute__((ext_vector_type(8))) float v8f;

// ---------------------------------------------------------------------------
// K1: keys = keys_param * rsqrt(sum(keys_param^2, -1) + 1e-12)
// one block per row, 256 threads
// ---------------------------------------------------------------------------
__global__ void __launch_bounds__(256)
k_normalize(const float* __restrict__ kp, float* __restrict__ kn) {
  __shared__ float red[256];
  const int row = blockIdx.x;
  const float* src = kp + (size_t)row * DD;
  float s = 0.f;
  for (int j = threadIdx.x; j < DD; j += 256) { float v = src[j]; s += v * v; }
  red[threadIdx.x] = s;
  __syncthreads();
  for (int off = 128; off > 0; off >>= 1) {
    if (threadIdx.x < off) red[threadIdx.x] += red[threadIdx.x + off];
    __syncthreads();
  }
  const float scale = rsqrtf(red[0] + 1e-12f);
  float* dst = kn + (size_t)row * DD;
  for (int j = threadIdx.x; j < DD; j += 256) dst[j] = src[j] * scale;
}

// ---------------------------------------------------------------------------
// K2: adj_sum[n1,n2] = sum_r adjacency[r,n1,n2]
// Streaming reduction: reads 201 MB, writes 16 MB -> the HBM-bound kernel.
// One float4 per thread, grid exactly covers N*N/4 elements.
// ---------------------------------------------------------------------------
__global__ void __launch_bounds__(256)
k_adj_reduce(const float* __restrict__ adj, float* __restrict__ adj_sum) {
  const size_t i4 = (size_t)blockIdx.x * 256 + threadIdx.x;   // float4 index
  const size_t stride4 = (size_t)NN * NN / 4;
  const v4f* a = (const v4f*)adj;
  v4f acc = a[i4];
#pragma unroll
  for (int r = 1; r < RR; ++r) acc += a[i4 + (size_t)r * stride4];
  ((v4f*)adj_sum)[i4] = acc;
}

// ---------------------------------------------------------------------------
// K3: scores(MM x NN) = positions(MM x DD) . keys^T  via V_WMMA_F32_16X16X4_F32
// One 16x16 C tile per wave, K stepped by 4 (192 WMMAs / tile).
// A frag (16x4 f32, 2 VGPR): lane<16 -> K0,K1 ; lane>=16 -> K2,K3 (row = lane&15)
// B frag mirrors A: B[kk][n] = keys[n][kk] -> contiguous float2 from keys row n.
// C/D (8 VGPR): VGPR v -> M = v + 8*(lane>=16), N = lane&15.
// ---------------------------------------------------------------------------
__global__ void __launch_bounds__(128)
k_gemm_scores(const float* __restrict__ A,   // MM x DD
              const float* __restrict__ Kn,  // NN x DD (normalized keys)
              float* __restrict__ S) {       // MM x NN
  const int wave = blockIdx.x * 4 + (threadIdx.x >> 5);
  const int lane = threadIdx.x & 31;
  const int tilesN = NN / 16;                // 128
  const int tr = wave / tilesN;              // 0..15
  const int tc = wave % tilesN;              // 0..127
  const int m  = lane & 15;
  const int kh = (lane >> 4) << 1;           // 0 or 2
  const float* arow = A  + (size_t)(tr * 16 + m) * DD;
  const float* brow = Kn + (size_t)(tc * 16 + m) * DD;  // n = tc*16 + (lane&15)
  v8f c = {};
#pragma unroll 8
  for (int k = 0; k < DD; k += 4) {
    v2f a = *(const v2f*)(arow + k + kh);
    v2f b = *(const v2f*)(brow + k + kh);
    c = __builtin_amdgcn_wmma_f32_16x16x4_f32(false, a, false, b,
                                              (short)0, c, false, false);
  }
  const int row0 = tr * 16 + ((lane >> 4) << 3);
  float* out = S + (size_t)row0 * NN + (tc * 16 + m);
#pragma unroll
  for (int v = 0; v < 8; ++v) out[(size_t)v * NN] = c[v];
}

// ---------------------------------------------------------------------------
// K4: per-row top-16 (iterative block argmax, tie-break on lower index to
// match lax.top_k) + softmax over the 16 values. One block per row.
// ---------------------------------------------------------------------------
__global__ void __launch_bounds__(256)
k_topk_softmax(const float* __restrict__ S, float* __restrict__ W,
               int* __restrict__ I) {
  __shared__ float sv[NN];
  __shared__ float rv[256];
  __shared__ int   ri[256];
  __shared__ float topv[KK];
  __shared__ int   topi[KK];
  const int row = blockIdx.x, tid = threadIdx.x;
  const float* src = S + (size_t)row * NN;
  for (int j = tid; j < NN; j += 256) sv[j] = src[j];
  __syncthreads();
  const float ninf = -__builtin_inff();
  for (int t = 0; t < KK; ++t) {
    float bv = ninf; int bi = NN;
    for (int j = tid; j < NN; j += 256) {
      float v = sv[j];
      if (v > bv || (v == bv && j < bi)) { bv = v; bi = j; }
    }
    rv[tid] = bv; ri[tid] = bi;
    __syncthreads();
    for (int off = 128; off > 0; off >>= 1) {
      if (tid < off) {
        float ov = rv[tid + off]; int oi = ri[tid + off];
        if (ov > rv[tid] || (ov == rv[tid] && oi < ri[tid])) {
          rv[tid] = ov; ri[tid] = oi;
        }
      }
      __syncthreads();
    }
    if (tid == 0) { topv[t] = rv[0]; topi[t] = ri[0]; sv[ri[0]] = ninf; }
    __syncthreads();
  }
  if (tid == 0) {
    const float mx = topv[0];            // selected in descending order
    float e[KK], sum = 0.f;
    for (int t = 0; t < KK; ++t) { e[t] = expf(topv[t] - mx); sum += e[t]; }
    const float inv = 1.0f / sum;
    for (int t = 0; t < KK; ++t) {
      W[row * KK + t] = e[t] * inv;
      I[row * KK + t] = topi[t];
    }
  }
}

// ---------------------------------------------------------------------------
// K5: nei[row, n] = sum_k w[row,k] * adj_sum[idx[row,k], n]
// 2 blocks per row, one float4 column slice per thread; adj_sum (16 MB) lives
// in the 192 MB L2, so the 33.5 MB of gathered rows mostly hit L2.
// ---------------------------------------------------------------------------
__global__ void __launch_bounds__(256)
k_nei_gather(const float* __restrict__ adj_sum, const float* __restrict__ W,
             const int* __restrict__ I, float* __restrict__ nei) {
  __shared__ float w[KK];
  __shared__ int   id[KK];
  const int row = blockIdx.x >> 1;
  const int n   = (((blockIdx.x & 1) << 8) + threadIdx.x) << 2;  // float4 col
  if (threadIdx.x < KK) {
    w[threadIdx.x]  = W[row * KK + threadIdx.x];
    id[threadIdx.x] = I[row * KK + threadIdx.x];
  }
  __syncthreads();
  v4f acc = {};
#pragma unroll
  for (int t = 0; t < KK; ++t) {
    acc += w[t] * *(const v4f*)(adj_sum + (size_t)id[t] * NN + n);
  }
  *(v4f*)(nei + (size_t)row * NN + n) = acc;
}

// ---------------------------------------------------------------------------
// K6: out(MM x DD) = nei(MM x NN) . keys(NN x DD) via V_WMMA_F32_16X16X4_F32
// B here is keys itself (K = n dim): b.x/b.y are stride-DD scalar loads;
// prefetch the next k-slice of B to hide L2 latency on the strided stream.
// ---------------------------------------------------------------------------
__global__ void __launch_bounds__(128)
k_gemm_out(const float* __restrict__ nei,  // MM x NN
           const float* __restrict__ Kn,   // NN x DD
           float* __restrict__ out) {      // MM x DD
  const int wave = blockIdx.x * 4 + (threadIdx.x >> 5);
  const int lane = threadIdx.x & 31;
  const int tilesN = DD / 16;               // 48
  const int tr = wave / tilesN;             // 0..15
  const int tc = wave % tilesN;             // 0..47
  const int m  = lane & 15;
  const int kh = (lane >> 4) << 1;
  const float* arow = nei + (size_t)(tr * 16 + m) * NN;
  const int col = tc * 16 + m;
  v8f c = {};
#pragma unroll 4
  for (int k = 0; k < NN; k += 4) {
    v2f a = *(const v2f*)(arow + k + kh);
    v2f b;
    b.x = Kn[(size_t)(k + kh)     * DD + col];
    b.y = Kn[(size_t)(k + kh + 1) * DD + col];
    if (k + 4 < NN)
      __builtin_prefetch(Kn + (size_t)(k + 4 + kh) * DD + col, 0, 3);
    c = __builtin_amdgcn_wmma_f32_16x16x4_f32(false, a, false, b,
                                              (short)0, c, false, false);
  }
  const int row0 = tr * 16 + ((lane >> 4) << 3);
  float* o = out + (size_t)row0 * DD + col;
#pragma unroll
  for (int v = 0; v < 8; ++v) o[(size_t)v * DD] = c[v];
}

// ---------------------------------------------------------------------------
// Host-side launcher. Inputs: positions (B,P,D) f32, keys_param (N,D) f32,
// adjacency (R,N,N) f32, k (==16). Output: (B,P,D) f32.
// Workspace layout (floats): keys_n | adj_sum | scores | nei | W | I  (~27 MB)
// ---------------------------------------------------------------------------
extern "C" void kernel_launch(void* const* d_in, const int* in_sizes, int n_in,
                              void* d_out, int out_size, void* d_ws,
                              size_t ws_size, hipStream_t stream) {
  const float* positions  = (const float*)d_in[0];
  const float* keys_param = (const float*)d_in[1];
  const float* adjacency  = (const float*)d_in[2];
  (void)in_sizes; (void)n_in; (void)out_size; (void)ws_size;

  float* ws = (float*)d_ws;
  size_t o = 0;
  float* keys_n  = ws + o;  o += (size_t)NN * DD;   // 1.5M floats
  float* adj_sum = ws + o;  o += (size_t)NN * NN;   // 4.0M floats
  float* scores  = ws + o;  o += (size_t)MM * NN;   // 0.5M floats
  float* nei     = ws + o;  o += (size_t)MM * NN;   // 0.5M floats
  float* W       = ws + o;  o += (size_t)MM * KK;
  int*   I       = (int*)(ws + o);

  // K1: row-normalize keys
  k_normalize<<<dim3(NN), dim3(256), 0, stream>>>(keys_param, keys_n);

  // K2: reduce adjacency over r (HBM-bound: 201 MB read)
  k_adj_reduce<<<dim3((NN * (size_t)NN) / 4 / 256), dim3(256), 0, stream>>>(
      adjacency, adj_sum);

  // K3: scores = positions . keys^T  (2048 waves, 4 waves/block)
  k_gemm_scores<<<dim3((MM / 16) * (NN / 16) / 4), dim3(128), 0, stream>>>(
      positions, keys_n, scores);

  // K4: top-16 + softmax per row
  k_topk_softmax<<<dim3(MM), dim3(256), 0, stream>>>(scores, W, I);

  // K5: nei = weighted gather of adj_sum rows
  k_nei_gather<<<dim3(MM * 2), dim3(256), 0, stream>>>(adj_sum, W, I, nei);

  // K6: out = nei . keys  (768 waves, 4 waves/block)
  k_gemm_out<<<dim3((MM / 16) * (DD / 16) / 4), dim3(128), 0, stream>>>(
      nei, keys_n, (float*)d_out);
}